// EGNNLayer_73272142070201
// MI455X (gfx1250) — compile-verified
//
#include <hip/hip_runtime.h>

typedef __attribute__((ext_vector_type(2))) float v2f;
typedef __attribute__((ext_vector_type(8))) float v8f;

#define TILE 64      // rows (edges or nodes) per workgroup
#define DD 128       // feature dim
#define KE 264       // edge-MLP K (261 padded to 264)
#define EIN_STR 268  // LDS stride for edge-input tile
#define H_STR 132    // LDS stride for hidden / ef tiles
#define AN_STR 260   // LDS stride for node-input tile (K=256)

// ---------------------------------------------------------------- zero ws
__global__ void egnn_zero_kernel(float* __restrict__ ws, long total) {
  long i = ((long)blockIdx.x * blockDim.x + threadIdx.x) * 4;
  if (i + 3 < total) {
    *(float4*)(ws + i) = make_float4(0.f, 0.f, 0.f, 0.f);
  } else {
    for (long j = i; j < total; ++j) ws[j] = 0.f;
  }
}

// ---------------------------------------------------------------- edge kernel
__global__ void __launch_bounds__(256)
egnn_edge_kernel(const float* __restrict__ x,
                 const long long* __restrict__ ei,   // [2,E]
                 const float* __restrict__ ea,       // [E,5]
                 const float* __restrict__ pos,      // [N,3]
                 const float* __restrict__ ew1, const float* __restrict__ eb1,
                 const float* __restrict__ ew2, const float* __restrict__ eb2,
                 const float* __restrict__ cw1, const float* __restrict__ cb1,
                 const float* __restrict__ cw2, const float* __restrict__ cb2,
                 float* __restrict__ agg,            // [N,128]
                 float* __restrict__ gacc,           // [3]
                 int E)
{
  extern __shared__ float smem[];
  float* Ein = smem;                     // [64][268]
  float* Hs  = smem + TILE * EIN_STR;    // [64][132]
  float* EF  = smem;                     // [64][132], reuses Ein after GEMM1

  __shared__ int   rowsI[TILE], colsI[TILE];
  __shared__ float cpart[TILE][4];
  __shared__ float sacc[3];

  const int t = threadIdx.x;
  const int ebase = blockIdx.x * TILE;

  if (t < TILE) {
    int e = ebase + t;
    if (e < E) { rowsI[t] = (int)ei[e]; colsI[t] = (int)ei[(long)E + e]; }
    else       { rowsI[t] = 0;          colsI[t] = 0; }
  }
  if (t < 3) sacc[t] = 0.f;
  __syncthreads();

  // ---- stage edge-input tile: [x[row] | x[col] | edge_attr | 0-pad]
  {
    const int r = t >> 2;            // 0..63
    const int q = t & 3;
    const int e = ebase + r;
    const bool ok = (e < E);
    const float* xr = x + (long)rowsI[r] * DD;
    const float* xc = x + (long)colsI[r] * DD;
    const float* ar = ea + (long)e * 5;
    float* dst = Ein + r * EIN_STR;
    #pragma unroll 1
    for (int j = 0; j < KE / 4; ++j) {
      int c = j * 4 + q;
      float v = 0.f;
      if (ok) {
        if (c < DD)          v = xr[c];
        else if (c < 2 * DD) v = xc[c - DD];
        else if (c < 2 * DD + 5) v = ar[c - 2 * DD];
      }
      dst[c] = v;
    }
  }
  __syncthreads();

  const int lane = t & 31;
  const int l15  = lane & 15;
  const int off  = (lane >> 4) << 1;          // 0 or 2 (K sub-offset)
  const int hi8  = (lane >> 4) << 3;          // 0 or 8 (C row offset)
  const int n    = (t >> 5) * 16 + l15;       // this lane's output column

  const v8f vzero = {};
  v8f acc[4];

  // ---- GEMM1: Ein[64][264] x ew1[261(pad 264)][128]
  #pragma unroll
  for (int m = 0; m < 4; ++m) acc[m] = vzero;
  {
    const float* w1n = ew1 + n;
    // main loop: K = 0..255 (64 chunks, power-of-two trip count)
    for (int kc = 0; kc < 64; ++kc) {
      const int kb = kc * 4 + off;
      v2f b;
      b.x = w1n[kb * DD];
      b.y = w1n[(kb + 1) * DD];
      #pragma unroll
      for (int m = 0; m < 4; ++m) {
        v2f a = *(const v2f*)&Ein[(m * 16 + l15) * EIN_STR + kb];
        acc[m] = __builtin_amdgcn_wmma_f32_16x16x4_f32(
            false, a, false, b, (short)0, acc[m], false, false);
      }
    }
    // peeled chunk kc=64: K = 256..259, all valid
    {
      const int kb = 256 + off;
      v2f b;
      b.x = w1n[kb * DD];
      b.y = w1n[(kb + 1) * DD];
      #pragma unroll
      for (int m = 0; m < 4; ++m) {
        v2f a = *(const v2f*)&Ein[(m * 16 + l15) * EIN_STR + kb];
        acc[m] = __builtin_amdgcn_wmma_f32_16x16x4_f32(
            false, a, false, b, (short)0, acc[m], false, false);
      }
    }
    // peeled chunk kc=65: K = 260..263, only K=260 is a real weight row
    {
      const int kb = 260 + off;
      v2f b;
      b.x = (off == 0) ? w1n[260 * DD] : 0.f;   // K=260 valid, 262 pad
      b.y = 0.f;                                // K=261 / 263 pad
      #pragma unroll
      for (int m = 0; m < 4; ++m) {
        v2f a = *(const v2f*)&Ein[(m * 16 + l15) * EIN_STR + kb];
        acc[m] = __builtin_amdgcn_wmma_f32_16x16x4_f32(
            false, a, false, b, (short)0, acc[m], false, false);
      }
    }
  }
  // bias + ReLU -> Hs
  {
    const float be = eb1[n];
    #pragma unroll
    for (int m = 0; m < 4; ++m)
      #pragma unroll
      for (int r = 0; r < 8; ++r) {
        float v = acc[m][r] + be;
        Hs[(m * 16 + r + hi8) * H_STR + n] = v > 0.f ? v : 0.f;
      }
  }
  __syncthreads();

  // ---- GEMM2: Hs[64][128] x ew2[128][128]
  #pragma unroll
  for (int m = 0; m < 4; ++m) acc[m] = vzero;
  {
    const float* w2n = ew2 + n;
    for (int kc = 0; kc < DD / 4; ++kc) {
      const int kb = kc * 4 + off;
      v2f b;
      b.x = w2n[kb * DD];
      b.y = w2n[(kb + 1) * DD];
      #pragma unroll
      for (int m = 0; m < 4; ++m) {
        v2f a = *(const v2f*)&Hs[(m * 16 + l15) * H_STR + kb];
        acc[m] = __builtin_amdgcn_wmma_f32_16x16x4_f32(
            false, a, false, b, (short)0, acc[m], false, false);
      }
    }
  }
  // bias -> EF (for coord MLP) + atomic scatter-add into agg[row]
  {
    const float be = eb2[n];
    #pragma unroll
    for (int m = 0; m < 4; ++m)
      #pragma unroll
      for (int r = 0; r < 8; ++r) {
        const int rowt = m * 16 + r + hi8;
        const float v = acc[m][r] + be;
        EF[rowt * H_STR + n] = v;
        if (ebase + rowt < E)
          atomicAdd(&agg[(long)rowsI[rowt] * DD + n], v);
      }
  }
  __syncthreads();

  // ---- coord MLP: w = relu(ef . cw1 + cb1)*cw2 + cb2 ; accumulate w*(pos[row]-pos[col])
  {
    const int e = t & 63;
    const int part = t >> 6;                  // 0..3, 32 cols each
    const float* c1  = cw1 + part * 32;
    const float* efr = EF + e * H_STR + part * 32;
    float s = 0.f;
    #pragma unroll
    for (int j = 0; j < 32; ++j) s += efr[j] * c1[j];
    cpart[e][part] = s;
  }
  __syncthreads();
  if (t < TILE && (ebase + t) < E) {
    float sum = cpart[t][0] + cpart[t][1] + cpart[t][2] + cpart[t][3] + cb1[0];
    float w = (sum > 0.f ? sum : 0.f) * cw2[0] + cb2[0];
    const int rn = rowsI[t], cn = colsI[t];
    atomicAdd(&sacc[0], w * (pos[rn * 3 + 0] - pos[cn * 3 + 0]));
    atomicAdd(&sacc[1], w * (pos[rn * 3 + 1] - pos[cn * 3 + 1]));
    atomicAdd(&sacc[2], w * (pos[rn * 3 + 2] - pos[cn * 3 + 2]));
  }
  __syncthreads();
  if (t < 3) atomicAdd(&gacc[t], sacc[t]);
}

// ---------------------------------------------------------------- node kernel
__global__ void __launch_bounds__(256)
egnn_node_kernel(const float* __restrict__ x,
                 const float* __restrict__ agg,
                 const float* __restrict__ nw1, const float* __restrict__ nb1,
                 const float* __restrict__ nw2, const float* __restrict__ nb2,
                 float* __restrict__ out_nf, int N)
{
  extern __shared__ float smem[];
  float* As = smem;                    // [64][260]
  float* Hs = smem + TILE * AN_STR;    // [64][132]

  const int t = threadIdx.x;
  const int nbase0 = blockIdx.x * TILE;

  // ---- stage node-input tile: [x | agg]
  {
    const int r = t >> 2;
    const int q = t & 3;
    const int node = nbase0 + r;
    const bool ok = (node < N);
    const float* xr = x   + (long)node * DD;
    const float* gr = agg + (long)node * DD;
    float* dst = As + r * AN_STR;
    #pragma unroll 1
    for (int j = 0; j < 256 / 4; ++j) {
      int c = j * 4 + q;
      float v = 0.f;
      if (ok) v = (c < DD) ? xr[c] : gr[c - DD];
      dst[c] = v;
    }
  }
  __syncthreads();

  const int lane = t & 31;
  const int l15  = lane & 15;
  const int off  = (lane >> 4) << 1;
  const int hi8  = (lane >> 4) << 3;
  const int n    = (t >> 5) * 16 + l15;

  const v8f vzero = {};
  v8f acc[4];

  // ---- GEMM1: As[64][256] x nw1[256][128]
  #pragma unroll
  for (int m = 0; m < 4; ++m) acc[m] = vzero;
  {
    const float* w1n = nw1 + n;
    for (int kc = 0; kc < 256 / 4; ++kc) {
      const int kb = kc * 4 + off;
      v2f b;
      b.x = w1n[kb * DD];
      b.y = w1n[(kb + 1) * DD];
      #pragma unroll
      for (int m = 0; m < 4; ++m) {
        v2f a = *(const v2f*)&As[(m * 16 + l15) * AN_STR + kb];
        acc[m] = __builtin_amdgcn_wmma_f32_16x16x4_f32(
            false, a, false, b, (short)0, acc[m], false, false);
      }
    }
  }
  {
    const float be = nb1[n];
    #pragma unroll
    for (int m = 0; m < 4; ++m)
      #pragma unroll
      for (int r = 0; r < 8; ++r) {
        float v = acc[m][r] + be;
        Hs[(m * 16 + r + hi8) * H_STR + n] = v > 0.f ? v : 0.f;
      }
  }
  __syncthreads();

  // ---- GEMM2: Hs[64][128] x nw2[128][128]
  #pragma unroll
  for (int m = 0; m < 4; ++m) acc[m] = vzero;
  {
    const float* w2n = nw2 + n;
    for (int kc = 0; kc < DD / 4; ++kc) {
      const int kb = kc * 4 + off;
      v2f b;
      b.x = w2n[kb * DD];
      b.y = w2n[(kb + 1) * DD];
      #pragma unroll
      for (int m = 0; m < 4; ++m) {
        v2f a = *(const v2f*)&Hs[(m * 16 + l15) * H_STR + kb];
        acc[m] = __builtin_amdgcn_wmma_f32_16x16x4_f32(
            false, a, false, b, (short)0, acc[m], false, false);
      }
    }
  }
  {
    const float be = nb2[n];
    #pragma unroll
    for (int m = 0; m < 4; ++m)
      #pragma unroll
      for (int r = 0; r < 8; ++r) {
        const int rowt = m * 16 + r + hi8;
        const int node = nbase0 + rowt;
        if (node < N) out_nf[(long)node * DD + n] = acc[m][r] + be;
      }
  }
}

// ---------------------------------------------------------------- pos kernel
__global__ void egnn_pos_kernel(const float* __restrict__ pos,
                                const float* __restrict__ gacc,
                                float* __restrict__ out_pos, int total) {
  int i = blockIdx.x * blockDim.x + threadIdx.x;
  if (i < total) out_pos[i] = pos[i] + gacc[i % 3];
}

// ---------------------------------------------------------------- launch
extern "C" void kernel_launch(void* const* d_in, const int* in_sizes, int n_in,
                              void* d_out, int out_size, void* d_ws, size_t ws_size,
                              hipStream_t stream) {
  const float*     x    = (const float*)d_in[0];
  const long long* ei   = (const long long*)d_in[1];
  const float*     ea   = (const float*)d_in[2];
  const float*     pos  = (const float*)d_in[3];
  const float*     ew1  = (const float*)d_in[4];
  const float*     eb1  = (const float*)d_in[5];
  const float*     ew2  = (const float*)d_in[6];
  const float*     eb2  = (const float*)d_in[7];
  const float*     nw1  = (const float*)d_in[8];
  const float*     nb1  = (const float*)d_in[9];
  const float*     nw2  = (const float*)d_in[10];
  const float*     nb2  = (const float*)d_in[11];
  const float*     cw1  = (const float*)d_in[12];
  const float*     cb1  = (const float*)d_in[13];
  const float*     cw2  = (const float*)d_in[14];
  const float*     cb2  = (const float*)d_in[15];

  const int N = in_sizes[0] / DD;
  const int E = in_sizes[2] / 5;

  float* wsf  = (float*)d_ws;
  float* gacc = wsf;            // [3] (+1 pad)
  float* agg  = wsf + 4;        // [N,128]
  float* out_nf  = (float*)d_out;
  float* out_pos = out_nf + (long)N * DD;

  // 1) zero agg + coord accumulator
  const long ztotal = 4 + (long)N * DD;
  const int  zgrid  = (int)((ztotal / 4 + 255) / 256) + 1;
  egnn_zero_kernel<<<zgrid, 256, 0, stream>>>(wsf, ztotal);

  // 2) edge MLP + scatter-add + coord reduction
  const int egrid = (E + TILE - 1) / TILE;
  const size_t elds = (size_t)(TILE * EIN_STR + TILE * H_STR) * sizeof(float);
  egnn_edge_kernel<<<egrid, 256, elds, stream>>>(
      x, ei, ea, pos, ew1, eb1, ew2, eb2, cw1, cb1, cw2, cb2, agg, gacc, E);

  // 3) node MLP
  const int ngrid = (N + TILE - 1) / TILE;
  const size_t nlds = (size_t)(TILE * AN_STR + TILE * H_STR) * sizeof(float);
  egnn_node_kernel<<<ngrid, 256, nlds, stream>>>(
      x, agg, nw1, nb1, nw2, nb2, out_nf, N);

  // 4) pos update
  const int ptotal = N * 3;
  egnn_pos_kernel<<<(ptotal + 255) / 256, 256, 0, stream>>>(pos, gacc, out_pos, ptotal);
}